// RankNetLoss_3427383902896
// MI455X (gfx1250) — compile-verified
//
#include <hip/hip_runtime.h>

// RankNet listwise loss: B=32768 rows x N=32 items, all-pairs BCE per row,
// mean over rows with valid pairs. One wave32 per row; LDS-staged row data
// via gfx1250 async global->LDS; f32 WMMA (16x16x4) for the lane reduction.

typedef __attribute__((ext_vector_type(2))) float v2f;
typedef __attribute__((ext_vector_type(8))) float v8f;

#define EPS_F 1e-8f

#define AS1 __attribute__((address_space(1)))
#define AS3 __attribute__((address_space(3)))

#if defined(__has_builtin)
#if __has_builtin(__builtin_amdgcn_global_load_async_to_lds_b32)
#define HAVE_ASYNC_LDS 1
#endif
#endif

__global__ __launch_bounds__(256) void ranknet_rows_kernel(
    const float* __restrict__ scores,
    const int*   __restrict__ rankings,
    const int*   __restrict__ mask,
    float*       __restrict__ ws,   // ws[0]=sum of per-row losses, ws[1]=rows with pairs
    int rows)
{
    __shared__ float s_sc[256];
    __shared__ int   s_rk[256];
    __shared__ int   s_mk[256];
    __shared__ int   s_vr[256];
    __shared__ float s_red[8][2];

    const int t    = threadIdx.x;
    const int lane = t & 31;
    const int wave = t >> 5;
    const int row  = blockIdx.x * 8 + wave;
    if (row >= rows) return;

    const int    widx = wave * 32;
    const size_t gidx = (size_t)row * 32 + (size_t)lane;

#if HAVE_ASYNC_LDS
    // gfx1250 async global -> LDS staging (tracked by ASYNCcnt)
    __builtin_amdgcn_global_load_async_to_lds_b32(
        (AS1 int*)(void*)(scores + gidx),   (AS3 int*)(void*)&s_sc[widx + lane], 0, 0);
    __builtin_amdgcn_global_load_async_to_lds_b32(
        (AS1 int*)(void*)(rankings + gidx), (AS3 int*)(void*)&s_rk[widx + lane], 0, 0);
    __builtin_amdgcn_global_load_async_to_lds_b32(
        (AS1 int*)(void*)(mask + gidx),     (AS3 int*)(void*)&s_mk[widx + lane], 0, 0);
#if __has_builtin(__builtin_amdgcn_s_wait_asynccnt)
    __builtin_amdgcn_s_wait_asynccnt(0);
#else
    asm volatile("s_wait_asynccnt 0" ::: "memory");
#endif
#else
    s_sc[widx + lane] = scores[gidx];
    s_rk[widx + lane] = rankings[gidx];
    s_mk[widx + lane] = mask[gidx];
#endif

    // Per-lane item j = lane of this wave's row.
    const float sj = s_sc[widx + lane];
    int vrj;
    {
        const int rk = s_rk[widx + lane];
        const int mk = s_mk[widx + lane];
        vrj = (mk != 0 && rk > 0) ? rk : -1;   // -1 == invalid
    }
    s_vr[widx + lane] = vrj;                   // same-wave LDS, in-order

    const bool vj = (vrj >= 0);
    float acc = 0.0f;   // sum of pair losses where j participates as 2nd index
    float cnt = 0.0f;   // number of valid ordered pairs (i, j=lane)

    // TEMPERATURE == 1.0 -> no scaling needed.
#pragma unroll 8
    for (int k = 0; k < 32; ++k) {
        const float si  = s_sc[widx + k];      // LDS broadcast read
        const int   vri = s_vr[widx + k];
        const bool  pv  = vj && (vri >= 0) && (vri != vrj);
        const float d   = si - sj;                             // s_i - s_j
        const float e   = __expf(-d);                          // v_exp_f32
        const float sig = __builtin_amdgcn_rcpf(1.0f + e);     // v_rcp_f32
        // p_ij = (r_i < r_j): loss = -log(sig+eps); else -log(1-sig+eps)
        const float tt   = (vri < vrj) ? sig : (1.0f - sig);
        const float loss = -__logf(tt + EPS_F);                // v_log_f32
        if (pv) { acc += loss; cnt += 1.0f; }
    }

    // ---- 32-lane reduction of (acc, cnt) with one f32 WMMA ----
    // A (16x4): lanes 0-15 hold {K0=acc, K1=cnt}, lanes 16-31 hold {K2=acc, K3=cnt}.
    // B (4x16): columns 0-7 select the acc K-slots, columns 8-15 the cnt K-slots.
    // => D[m][n<8]  = acc[m] + acc[m+16],  D[m][n>=8] = cnt[m] + cnt[m+16]
    v2f a = { acc, cnt };
    const float bx = ((lane & 15) < 8) ? 1.0f : 0.0f;
    v2f b = { bx, 1.0f - bx };
    v8f c = {};
    c = __builtin_amdgcn_wmma_f32_16x16x4_f32(
            /*neg_a=*/false, a, /*neg_b=*/false, b,
            /*c_mod=*/(short)0, c, /*reuse_a=*/false, /*reuse_b=*/false);

    // Per-lane column sum over the 8 C/D VGPRs (M=0..7 or M=8..15), then fold halves.
    float tot = c[0] + c[1] + c[2] + c[3] + c[4] + c[5] + c[6] + c[7];
    tot += __shfl_xor(tot, 16, 32);
    const float accT = __shfl(tot, 0, 32);   // any column 0-7 carries total acc
    const float cntT = __shfl(tot, 8, 32);   // any column 8-15 carries total cnt

    if (lane == 0) {
        const bool has = (cntT > 0.5f);
        s_red[wave][0] = has ? (accT / cntT) : 0.0f;
        s_red[wave][1] = has ? 1.0f : 0.0f;
    }
    __syncthreads();

    if (t == 0) {
        float ls = 0.0f, hs = 0.0f;
#pragma unroll
        for (int w = 0; w < 8; ++w) { ls += s_red[w][0]; hs += s_red[w][1]; }
        atomicAdd(&ws[0], ls);
        atomicAdd(&ws[1], hs);
    }
}

__global__ void ranknet_init_kernel(float* ws)
{
    if (threadIdx.x == 0) { ws[0] = 0.0f; ws[1] = 0.0f; }
}

__global__ void ranknet_final_kernel(const float* __restrict__ ws,
                                     float* __restrict__ out)
{
    if (threadIdx.x == 0) {
        const float n = ws[1];
        out[0] = (n > 0.0f) ? (ws[0] / fmaxf(n, 1.0f)) : 0.0f;
    }
}

extern "C" void kernel_launch(void* const* d_in, const int* in_sizes, int n_in,
                              void* d_out, int out_size, void* d_ws, size_t ws_size,
                              hipStream_t stream)
{
    const float* scores   = (const float*)d_in[0];
    const int*   rankings = (const int*)d_in[1];
    const int*   mask     = (const int*)d_in[2];
    float*       out      = (float*)d_out;
    float*       ws       = (float*)d_ws;

    const int rows   = in_sizes[0] / 32;          // 32768
    const int blocks = (rows + 7) / 8;            // 8 rows (waves) per block

    hipLaunchKernelGGL(ranknet_init_kernel, dim3(1), dim3(32), 0, stream, ws);
    hipLaunchKernelGGL(ranknet_rows_kernel, dim3(blocks), dim3(256), 0, stream,
                       scores, rankings, mask, ws, rows);
    hipLaunchKernelGGL(ranknet_final_kernel, dim3(1), dim3(32), 0, stream, ws, out);
}